// ClusterLocalAtt_89163521065538
// MI455X (gfx1250) — compile-verified
//
#include <hip/hip_runtime.h>
#include <math.h>

// ---------------- problem constants ----------------
#define CCH   256          // channels
#define HH    512
#define WW    512
#define HWF   262144       // H*W
#define HD    128
#define WD    128
#define NHD   16384        // HD*WD
#define NNOI  32
#define NOFF  224          // first noise channel (noise_idx = 224..255)

typedef float v2f __attribute__((ext_vector_type(2)));
typedef float v8f __attribute__((ext_vector_type(8)));

#define V8F_ZERO {0.f,0.f,0.f,0.f,0.f,0.f,0.f,0.f}

// ---------------------------------------------------------------------------
// K1: 4x4 max-pool  x(C,512,512) -> xd(C,128,128)
// grid = C*HD blocks, 128 threads (one output per thread, float4 row loads)
// ---------------------------------------------------------------------------
__global__ void k_pool(const float* __restrict__ x, float* __restrict__ xd) {
  int c  = blockIdx.x >> 7;
  int hd = blockIdx.x & 127;
  int wd = threadIdx.x;
  const float* base = x + (size_t)c * HWF + (size_t)(hd * 4) * WW + wd * 4;
  float m = -3.402823466e38f;
#pragma unroll
  for (int r = 0; r < 4; ++r) {
    const float4 v = *(const float4*)(base + r * WW);
    m = fmaxf(m, fmaxf(fmaxf(v.x, v.y), fmaxf(v.z, v.w)));
  }
  xd[(size_t)c * NHD + hd * WD + wd] = m;
}

// ---------------------------------------------------------------------------
// K2: per-channel row sum of xd -> R[c]   (grid=256 blocks, 256 thr)
// ---------------------------------------------------------------------------
__global__ void k_rowsum(const float* __restrict__ xd, float* __restrict__ R) {
  __shared__ float sm[256];
  int c = blockIdx.x, t = threadIdx.x;
  float s = 0.f;
  for (int i = t; i < NHD; i += 256) s += xd[(size_t)c * NHD + i];
  sm[t] = s; __syncthreads();
  for (int off = 128; off > 0; off >>= 1) {
    if (t < off) sm[t] += sm[t + off];
    __syncthreads();
  }
  if (t == 0) R[c] = sm[0];
}

// ---------------------------------------------------------------------------
// K3: Gram matrix G = xd * xd^T  (256x256, K=16384) via V_WMMA_F32_16X16X4_F32
// grid = 32 blocks * 8 waves = 256 tiles of 16x16
// ---------------------------------------------------------------------------
__global__ void k_gram(const float* __restrict__ xd, float* __restrict__ G) {
  int wave = threadIdx.x >> 5, lane = threadIdx.x & 31;
  int t  = blockIdx.x * 8 + wave;     // 0..255
  int ti = t >> 4, tj = t & 15;
  int l15  = lane & 15;
  int koff = (lane >> 4) * 2;
  const float* Arow = xd + (size_t)(ti * 16 + l15) * NHD + koff;
  const float* Brow = xd + (size_t)(tj * 16 + l15) * NHD + koff;
  v8f acc = V8F_ZERO;
#pragma unroll 8
  for (int k = 0; k < NHD; k += 4) {
    v2f a = *(const v2f*)(Arow + k);
    v2f b = *(const v2f*)(Brow + k);
    acc = __builtin_amdgcn_wmma_f32_16x16x4_f32(false, a, false, b,
                                                (short)0, acc, false, false);
  }
  int mb = ti * 16 + 8 * (lane >> 4);
#pragma unroll
  for (int r = 0; r < 8; ++r)
    G[(size_t)(mb + r) * CCH + tj * 16 + l15] = acc[r];
}

// ---------------------------------------------------------------------------
// K4: masked softmax over the affine-decomposed scores; emit
//     Aprime[i,j] = attn[i,j]*w_qkv[512+j]  and beta[i] = sum attn*b_qkv[512+j]
// single block of 256 threads (row per thread)
// ---------------------------------------------------------------------------
__global__ void k_attn(const float* __restrict__ G, const float* __restrict__ R,
                       const int* __restrict__ labels,
                       const float* __restrict__ wq, const float* __restrict__ bq,
                       float* __restrict__ Ap, float* __restrict__ beta) {
  __shared__ int sl[256];
  int i = threadIdx.x;
  sl[i] = labels[i];
  __syncthreads();
  int li = sl[i];
  if (li == -1) {                       // invalid row -> attn row zeroed
    for (int j = 0; j < CCH; ++j) Ap[(size_t)i * CCH + j] = 0.f;
    beta[i] = 0.f;
    return;
  }
  float ai = wq[i], bi = bq[i];
  int   qi = i / 3;
  float Rq = R[qi];
  const float* Gq = G + (size_t)qi * CCH;

  float m = -3.402823466e38f;
  for (int j = 0; j < CCH; ++j) {
    if (sl[j] == li) {
      int kj = (256 + j) / 3;
      float aj = wq[256 + j], bj = bq[256 + j];
      float s = ai * aj * Gq[kj] + ai * bj * Rq + bi * aj * R[kj]
                + 16384.f * bi * bj;
      m = fmaxf(m, s);
    }
  }
  float den = 0.f;
  for (int j = 0; j < CCH; ++j) {
    if (sl[j] == li) {
      int kj = (256 + j) / 3;
      float aj = wq[256 + j], bj = bq[256 + j];
      float s = ai * aj * Gq[kj] + ai * bj * Rq + bi * aj * R[kj]
                + 16384.f * bi * bj;
      den += expf(s - m);
    }
  }
  float inv = 1.f / den;
  float bsum = 0.f;
  for (int j = 0; j < CCH; ++j) {
    float p = 0.f;
    if (sl[j] == li) {
      int kj = (256 + j) / 3;
      float aj = wq[256 + j], bj = bq[256 + j];
      float s = ai * aj * Gq[kj] + ai * bj * Rq + bi * aj * R[kj]
                + 16384.f * bi * bj;
      p = expf(s - m) * inv;
    }
    Ap[(size_t)i * CCH + j] = p * wq[512 + j];
    bsum += p * bq[512 + j];
  }
  beta[i] = bsum;
}

// ---------------------------------------------------------------------------
// K5: d = Aprime @ xd[vsrc(j)] + beta   (256 x 16384, K=256) via WMMA f32
// grid = 2048 blocks * 8 waves = 16384 tiles (16 Mtiles x 1024 Ntiles)
// ---------------------------------------------------------------------------
__global__ void k_outds(const float* __restrict__ Ap, const float* __restrict__ xd,
                        const float* __restrict__ beta, float* __restrict__ d) {
  int wave = threadIdx.x >> 5, lane = threadIdx.x & 31;
  int t  = blockIdx.x * 8 + wave;        // 0..16383
  int mt = t >> 10;                      // 0..15
  int nt = t & 1023;                     // 0..1023
  int l15  = lane & 15;
  int koff = (lane >> 4) * 2;
  int n0 = nt * 16;
  const float* Arow = Ap + (size_t)(mt * 16 + l15) * CCH + koff;
  v8f acc = V8F_ZERO;
#pragma unroll 4
  for (int k = 0; k < CCH; k += 4) {
    v2f a = *(const v2f*)(Arow + k);
    int k0 = k + koff;
    int v0 = (512 + k0) / 3;             // v-row gather: channel source of v[j]
    int v1 = (513 + k0) / 3;
    v2f b;
    b.x = xd[(size_t)v0 * NHD + n0 + l15];
    b.y = xd[(size_t)v1 * NHD + n0 + l15];
    acc = __builtin_amdgcn_wmma_f32_16x16x4_f32(false, a, false, b,
                                                (short)0, acc, false, false);
  }
  int mb = mt * 16 + 8 * (lane >> 4);
#pragma unroll
  for (int r = 0; r < 8; ++r) {
    int row = mb + r;
    d[(size_t)row * NHD + n0 + l15] = acc[r] + beta[row];
  }
}

// ---------------------------------------------------------------------------
// K6: per-channel sum(d), sum(d^2)  (for closed-form ||up[c]||)
// ---------------------------------------------------------------------------
__global__ void k_dsum(const float* __restrict__ d, float* __restrict__ sd,
                       float* __restrict__ sd2) {
  __shared__ float s1[256], s2[256];
  int c = blockIdx.x, t = threadIdx.x;
  float a = 0.f, b = 0.f;
  for (int i = t; i < NHD; i += 256) {
    float v = d[(size_t)c * NHD + i];
    a += v; b += v * v;
  }
  s1[t] = a; s2[t] = b; __syncthreads();
  for (int off = 128; off > 0; off >>= 1) {
    if (t < off) { s1[t] += s1[t + off]; s2[t] += s2[t + off]; }
    __syncthreads();
  }
  if (t == 0) { sd[c] = s1[0]; sd2[c] = s2[0]; }
}

// ---------------------------------------------------------------------------
// K7: dot(up[c], x_noise[n]) as a K-split WMMA GEMM (M=256, N=32, K=262144)
// with the `up` operand reconstructed on the fly from d, w_up, b_up
// (up is never materialized).  grid = 256 blocks * 8 waves = 2048 waves:
// 16 Mtiles x 2 Ntiles x 64 K-splits; each wave does 1024 wmma steps.
// ---------------------------------------------------------------------------
__global__ void k_dotp(const float* __restrict__ d, const float* __restrict__ x,
                       const int* __restrict__ noise_idx,
                       const float* __restrict__ w_up, const float* __restrict__ b_up,
                       float* __restrict__ part) {
  int wave = threadIdx.x >> 5, lane = threadIdx.x & 31;
  int t  = blockIdx.x * 8 + wave;        // 0..2047
  int ks = t & 63;
  int nt = (t >> 6) & 1;
  int mt = t >> 7;                       // 0..15
  int l15  = lane & 15;
  int koff = (lane >> 4) * 2;
  int c   = mt * 16 + l15;               // A-row channel
  int nch = noise_idx[nt * 16 + l15];    // B-col noise channel
  float bup = b_up[c];
  // per-lane upsample weights for its koff pair, one per row-phase a=0..3
  float wv0[4], wv1[4];
#pragma unroll
  for (int aa = 0; aa < 4; ++aa) {
    wv0[aa] = w_up[c * 16 + aa * 4 + koff];
    wv1[aa] = w_up[c * 16 + aa * 4 + koff + 1];
  }
  const float* drowc = d + (size_t)c * NHD;
  const float* xrow  = x + (size_t)nch * HWF + koff;
  v8f acc = V8F_ZERO;
  int hh0 = ks * 8;                      // hh0 % 4 == 0 -> phase starts at 0
#pragma unroll
  for (int r = 0; r < 8; ++r) {
    int hh = hh0 + r;
    float w0 = wv0[r & 3];
    float w1 = wv1[r & 3];
    const float* dpr = drowc + ((hh >> 2) << 7);     // 128-entry d row
    const float* xpr = xrow + (size_t)hh * WW;
#pragma unroll 4
    for (int j = 0; j < 128; ++j) {
      float dv = dpr[j];
      v2f a; a.x = dv * w0 + bup; a.y = dv * w1 + bup;
      v2f b = *(const v2f*)(xpr + 4 * j);
      acc = __builtin_amdgcn_wmma_f32_16x16x4_f32(false, a, false, b,
                                                  (short)0, acc, false, false);
    }
  }
  int base = (((mt << 1) | nt) * 64 + ks) * 256;
#pragma unroll
  for (int r = 0; r < 8; ++r)
    part[base + (r + 8 * (lane >> 4)) * 16 + l15] = acc[r];
}

// K8: reduce k-split partials -> dot[c*32+n]   (8192 outputs)
__global__ void k_dotred(const float* __restrict__ part, float* __restrict__ dotv) {
  int g = blockIdx.x * 256 + threadIdx.x;   // 0..8191
  int c = g >> 5, n = g & 31;
  int mt = c >> 4, ml = c & 15, nt = n >> 4, nl = n & 15;
  int base = (((mt << 1) | nt) * 64) * 256 + ml * 16 + nl;
  float s = 0.f;
  for (int ks = 0; ks < 64; ++ks) s += part[base + ks * 256];
  dotv[c * 32 + n] = s;
}

// ---------------------------------------------------------------------------
// K9: nf output copy (x noise channels) + ||x[n]||^2
// ---------------------------------------------------------------------------
__global__ void k_nf(const float* __restrict__ x, const int* __restrict__ noise_idx,
                     float* __restrict__ out2, float* __restrict__ xn2) {
  __shared__ float sm[256];
  int n = blockIdx.x, t = threadIdx.x;
  int ch = noise_idx[n];
  const float4* src = (const float4*)(x + (size_t)ch * HWF);
  float4* dst = (float4*)(out2 + (size_t)n * HWF);
  float s = 0.f;
  for (int i = t; i < HWF / 4; i += 256) {
    float4 v = src[i];
    dst[i] = v;
    s += v.x * v.x + v.y * v.y + v.z * v.z + v.w * v.w;
  }
  sm[t] = s; __syncthreads();
  for (int off = 128; off > 0; off >>= 1) {
    if (t < off) sm[t] += sm[t + off];
    __syncthreads();
  }
  if (t == 0) xn2[n] = sm[0];
}

// ---------------------------------------------------------------------------
// K10: coef[c] = 1 - mean_n dot/(||up||*||xn||), ||up|| in closed form
// ---------------------------------------------------------------------------
__global__ void k_sim(const float* __restrict__ w_up, const float* __restrict__ b_up,
                      const float* __restrict__ sd, const float* __restrict__ sd2,
                      const float* __restrict__ dotv, const float* __restrict__ xn2,
                      float* __restrict__ coef) {
  int c = threadIdx.x;
  float sw = 0.f, sw2 = 0.f;
#pragma unroll
  for (int i = 0; i < 16; ++i) {
    float w = w_up[c * 16 + i];
    sw += w; sw2 += w * w;
  }
  float bup = b_up[c];
  float n2 = sw2 * sd2[c] + 2.f * bup * sw * sd[c] + 262144.f * bup * bup;
  float nup = fmaxf(sqrtf(fmaxf(n2, 0.f)), 1e-12f);
  float s = 0.f;
  for (int n = 0; n < NNOI; ++n) {
    float nx = fmaxf(sqrtf(xn2[n]), 1e-12f);
    s += dotv[c * 32 + n] / (nup * nx);
  }
  coef[c] = 1.f - s * (1.f / 32.f);
}

// ---------------------------------------------------------------------------
// K11: out = up (recomputed from d) + coef[c]*x    grid = C*H blocks, 128 thr
// ---------------------------------------------------------------------------
__global__ void k_final(const float* __restrict__ d, const float* __restrict__ x,
                        const float* __restrict__ w_up, const float* __restrict__ b_up,
                        const float* __restrict__ coef, float* __restrict__ out) {
  int c  = blockIdx.x >> 9;
  int hh = blockIdx.x & 511;
  int aa = hh & 3;
  float w0 = w_up[c * 16 + aa * 4 + 0];
  float w1 = w_up[c * 16 + aa * 4 + 1];
  float w2 = w_up[c * 16 + aa * 4 + 2];
  float w3 = w_up[c * 16 + aa * 4 + 3];
  float bup = b_up[c];
  float cf  = coef[c];
  int wd = threadIdx.x;
  float dv = d[(size_t)c * NHD + (hh >> 2) * WD + wd];
  size_t off = (size_t)c * HWF + (size_t)hh * WW + wd * 4;
  const float4 xv = *(const float4*)(x + off);
  float4 o;
  o.x = dv * w0 + bup + cf * xv.x;
  o.y = dv * w1 + bup + cf * xv.y;
  o.z = dv * w2 + bup + cf * xv.z;
  o.w = dv * w3 + bup + cf * xv.w;
  *(float4*)(out + off) = o;
}

// ---------------------------------------------------------------------------
extern "C" void kernel_launch(void* const* d_in, const int* in_sizes, int n_in,
                              void* d_out, int out_size, void* d_ws, size_t ws_size,
                              hipStream_t stream) {
  const float* x       = (const float*)d_in[0];
  const int*   labels  = (const int*)d_in[1];
  const int*   nidx    = (const int*)d_in[2];
  const float* wq      = (const float*)d_in[3];
  const float* bq      = (const float*)d_in[4];
  const float* w_up    = (const float*)d_in[5];
  const float* b_up    = (const float*)d_in[6];
  float* out  = (float*)d_out;
  float* out2 = out + (size_t)CCH * HWF;      // nf tuple output

  float* ws = (float*)d_ws;
  float* xd      = ws;                 ws += (size_t)CCH * NHD;   // 4,194,304
  float* G       = ws;                 ws += CCH * CCH;           //    65,536
  float* R       = ws;                 ws += CCH;
  float* Ap      = ws;                 ws += CCH * CCH;
  float* beta    = ws;                 ws += CCH;
  float* dmat    = ws;                 ws += (size_t)CCH * NHD;   // 4,194,304
  float* sd      = ws;                 ws += CCH;
  float* sd2     = ws;                 ws += CCH;
  float* part    = ws;                 ws += 2048 * 256;          //   524,288
  float* dotv    = ws;                 ws += CCH * NNOI;
  float* xn2     = ws;                 ws += NNOI;
  float* coef    = ws;                 ws += CCH;

  k_pool  <<<CCH * HD, 128, 0, stream>>>(x, xd);
  k_rowsum<<<CCH, 256, 0, stream>>>(xd, R);
  k_gram  <<<32, 256, 0, stream>>>(xd, G);
  k_attn  <<<1, 256, 0, stream>>>(G, R, labels, wq, bq, Ap, beta);
  k_outds <<<2048, 256, 0, stream>>>(Ap, xd, beta, dmat);
  k_dsum  <<<CCH, 256, 0, stream>>>(dmat, sd, sd2);
  k_dotp  <<<256, 256, 0, stream>>>(dmat, x, nidx, w_up, b_up, part);
  k_dotred<<<32, 256, 0, stream>>>(part, dotv);
  k_nf    <<<NNOI, 256, 0, stream>>>(x, nidx, out2, xn2);
  k_sim   <<<1, 256, 0, stream>>>(w_up, b_up, sd, sd2, dotv, xn2, coef);
  k_final <<<CCH * HH, 128, 0, stream>>>(dmat, x, w_up, b_up, coef, out);
}